// GIN_71116068488095
// MI455X (gfx1250) — compile-verified
//
#include <hip/hip_runtime.h>
#include <hip/hip_bf16.h>
#include <math.h>

typedef float v8f __attribute__((ext_vector_type(8)));
typedef float v2f __attribute__((ext_vector_type(2)));

#define N_NODES   50000
#define N_EDGES   1600000
#define DIM       64
#define EDGE_DIM  32
#define N_GRAPHS  128
#define N_CLASSES 10

// ---------------------------------------------------------------------------
// zero-fill (workspace is poisoned 0xAA; must re-zero deterministically)
// ---------------------------------------------------------------------------
__global__ __launch_bounds__(256) void zero_kernel(float* __restrict__ p, int n) {
    int i = blockIdx.x * 256 + threadIdx.x;
    if (i < n) p[i] = 0.0f;
}

// ---------------------------------------------------------------------------
// Edge kernel: e = edge_attr @ We + be  (WMMA f32 16x16x4),
// msg = relu(x[src] + e), atomic scatter-add into aggr[dst].
// 8 waves/block, 16 edges per wave -> 128 edges per block.
// ---------------------------------------------------------------------------
__global__ __launch_bounds__(256) void edge_msg_kernel(
    const float* __restrict__ xin,    // [N,64]
    const float* __restrict__ eattr,  // [E,32]
    const float* __restrict__ We,     // [32,64]
    const float* __restrict__ be,     // [64]
    const int*   __restrict__ src,    // [E]
    const int*   __restrict__ dst,    // [E]
    float*       __restrict__ aggr,   // [N,64] pre-zeroed
    int E)
{
    __shared__ float sWe[EDGE_DIM * DIM];      // 8 KB
    __shared__ float sbe[DIM];
    __shared__ float sTile[8][16 * EDGE_DIM];  // 16 KB
    __shared__ int   sSrc[8][16];
    __shared__ int   sDst[8][16];

    const int tid  = threadIdx.x;
    const int wave = tid >> 5;
    const int lane = tid & 31;
    const int half = lane >> 4;   // 0: lanes 0-15, 1: lanes 16-31
    const int l15  = lane & 15;

    for (int i = tid; i < EDGE_DIM * DIM; i += 256) sWe[i] = We[i];
    if (tid < DIM) sbe[tid] = be[tid];
    __syncthreads();

    const int e0 = (blockIdx.x * 8 + wave) * 16;
    if (e0 >= E) return;

    // stage edge_attr tile [16 x 32] (row-major), coalesced
    for (int i = lane; i < 16 * EDGE_DIM; i += 32)
        sTile[wave][i] = eattr[(size_t)e0 * EDGE_DIM + i];
    if (lane < 16) {
        sSrc[wave][lane] = src[e0 + lane];
        sDst[wave][lane] = dst[e0 + lane];
    }

    // accumulators: 4 N-tiles of 16 cols; fold bias into C
    v8f c[4];
#pragma unroll
    for (int j = 0; j < 4; ++j)
#pragma unroll
        for (int r = 0; r < 8; ++r)
            c[j][r] = sbe[16 * j + l15];

    // e[16x64] = tile[16x32] @ We[32x64]; K in steps of 4
#pragma unroll
    for (int k0 = 0; k0 < EDGE_DIM; k0 += 4) {
        const int kr = k0 + 2 * half;
        v2f a;
        a.x = sTile[wave][l15 * EDGE_DIM + kr];
        a.y = sTile[wave][l15 * EDGE_DIM + kr + 1];
#pragma unroll
        for (int j = 0; j < 4; ++j) {
            v2f b;
            b.x = sWe[kr * DIM + 16 * j + l15];
            b.y = sWe[(kr + 1) * DIM + 16 * j + l15];
            c[j] = __builtin_amdgcn_wmma_f32_16x16x4_f32(
                false, a, false, b, (short)0, c[j], false, false);
        }
    }

    // C layout: c[j][r] = e[row = r + 8*half][col = 16*j + l15]
    // gather x[src], relu, scatter-add to aggr[dst]
#pragma unroll
    for (int r = 0; r < 8; ++r) {
        const int m = r + 8 * half;
        const int s = sSrc[wave][m];
        const int d = sDst[wave][m];
        const size_t srow = (size_t)s * DIM;
        const size_t drow = (size_t)d * DIM;
#pragma unroll
        for (int j = 0; j < 4; ++j) {
            float v = c[j][r] + xin[srow + 16 * j + l15];
            v = fmaxf(v, 0.0f);
            atomicAdd(&aggr[drow + 16 * j + l15], v);
        }
    }
}

// ---------------------------------------------------------------------------
// Node kernel: out = relu(((x + aggr) @ Wa + ba) @ Wb + bb)
// 4 waves/block (LDS budget), 16 nodes per wave.
// ---------------------------------------------------------------------------
__global__ __launch_bounds__(128) void node_mlp_kernel(
    const float* __restrict__ xin,   // [N,64]
    const float* __restrict__ aggr,  // [N,64]
    const float* __restrict__ Wa, const float* __restrict__ ba,
    const float* __restrict__ Wb, const float* __restrict__ bb,
    float*       __restrict__ out,   // [N,64]
    int N)
{
    __shared__ float sWa[DIM * DIM];   // 16 KB
    __shared__ float sWb[DIM * DIM];   // 16 KB
    __shared__ float sba[DIM], sbb[DIM];
    __shared__ float sT[4][16 * DIM];  // 16 KB

    const int tid  = threadIdx.x;
    const int wave = tid >> 5;
    const int lane = tid & 31;
    const int half = lane >> 4;
    const int l15  = lane & 15;

    for (int i = tid; i < DIM * DIM; i += 128) { sWa[i] = Wa[i]; sWb[i] = Wb[i]; }
    if (tid < DIM) { sba[tid] = ba[tid]; sbb[tid] = bb[tid]; }
    __syncthreads();

    const int n0 = (blockIdx.x * 4 + wave) * 16;
    if (n0 >= N) return;

    // stage t = x + aggr  [16 x 64], coalesced
    for (int i = lane; i < 16 * DIM; i += 32) {
        size_t g = (size_t)n0 * DIM + i;
        sT[wave][i] = xin[g] + aggr[g];
    }

    v8f c[4];

    // ---- u = t @ Wa + ba ----
#pragma unroll
    for (int j = 0; j < 4; ++j)
#pragma unroll
        for (int r = 0; r < 8; ++r)
            c[j][r] = sba[16 * j + l15];
#pragma unroll
    for (int k0 = 0; k0 < DIM; k0 += 4) {
        const int kr = k0 + 2 * half;
        v2f a;
        a.x = sT[wave][l15 * DIM + kr];
        a.y = sT[wave][l15 * DIM + kr + 1];
#pragma unroll
        for (int j = 0; j < 4; ++j) {
            v2f b;
            b.x = sWa[kr * DIM + 16 * j + l15];
            b.y = sWa[(kr + 1) * DIM + 16 * j + l15];
            c[j] = __builtin_amdgcn_wmma_f32_16x16x4_f32(
                false, a, false, b, (short)0, c[j], false, false);
        }
    }

    // write u back into sT (same-wave LDS ops stay in order; reads above
    // were issued before these stores)
#pragma unroll
    for (int j = 0; j < 4; ++j)
#pragma unroll
        for (int r = 0; r < 8; ++r)
            sT[wave][(r + 8 * half) * DIM + 16 * j + l15] = c[j][r];

    // ---- v = u @ Wb + bb ----
#pragma unroll
    for (int j = 0; j < 4; ++j)
#pragma unroll
        for (int r = 0; r < 8; ++r)
            c[j][r] = sbb[16 * j + l15];
#pragma unroll
    for (int k0 = 0; k0 < DIM; k0 += 4) {
        const int kr = k0 + 2 * half;
        v2f a;
        a.x = sT[wave][l15 * DIM + kr];
        a.y = sT[wave][l15 * DIM + kr + 1];
#pragma unroll
        for (int j = 0; j < 4; ++j) {
            v2f b;
            b.x = sWb[kr * DIM + 16 * j + l15];
            b.y = sWb[(kr + 1) * DIM + 16 * j + l15];
            c[j] = __builtin_amdgcn_wmma_f32_16x16x4_f32(
                false, a, false, b, (short)0, c[j], false, false);
        }
    }

    // relu + store
#pragma unroll
    for (int r = 0; r < 8; ++r) {
        const size_t row = (size_t)(n0 + r + 8 * half) * DIM;
#pragma unroll
        for (int j = 0; j < 4; ++j)
            out[row + 16 * j + l15] = fmaxf(c[j][r], 0.0f);
    }
}

// ---------------------------------------------------------------------------
// Global mean pool: atomic sums per graph + counts
// ---------------------------------------------------------------------------
__global__ __launch_bounds__(256) void pool_sum_kernel(
    const float* __restrict__ h,      // [N,64]
    const int*   __restrict__ batch,  // [N]
    float* __restrict__ sums,         // [G,64] pre-zeroed
    float* __restrict__ cnts,         // [G]    pre-zeroed
    int N)
{
    int idx = blockIdx.x * 256 + threadIdx.x;
    if (idx >= N * DIM) return;
    int node = idx >> 6;
    int col  = idx & 63;
    int g = batch[node];
    atomicAdd(&sums[(size_t)g * DIM + col], h[idx]);
    if (col == 0) atomicAdd(&cnts[g], 1.0f);
}

// ---------------------------------------------------------------------------
// Head: pooled mean -> linear -> softmax. One thread per graph.
// ---------------------------------------------------------------------------
__global__ __launch_bounds__(128) void head_kernel(
    const float* __restrict__ sums, const float* __restrict__ cnts,
    const float* __restrict__ Wf, const float* __restrict__ bf,
    float* __restrict__ out)
{
    int g = threadIdx.x;
    if (g >= N_GRAPHS) return;
    float inv = 1.0f / fmaxf(cnts[g], 1.0f);
    float logits[N_CLASSES];
#pragma unroll
    for (int j = 0; j < N_CLASSES; ++j) logits[j] = bf[j];
    for (int i = 0; i < DIM; ++i) {
        float p = sums[(size_t)g * DIM + i] * inv;
#pragma unroll
        for (int j = 0; j < N_CLASSES; ++j)
            logits[j] += p * Wf[i * N_CLASSES + j];
    }
    float mx = logits[0];
#pragma unroll
    for (int j = 1; j < N_CLASSES; ++j) mx = fmaxf(mx, logits[j]);
    float s = 0.0f;
#pragma unroll
    for (int j = 0; j < N_CLASSES; ++j) { logits[j] = __expf(logits[j] - mx); s += logits[j]; }
    float r = 1.0f / s;
#pragma unroll
    for (int j = 0; j < N_CLASSES; ++j) out[(size_t)g * N_CLASSES + j] = logits[j] * r;
}

// ---------------------------------------------------------------------------
extern "C" void kernel_launch(void* const* d_in, const int* in_sizes, int n_in,
                              void* d_out, int out_size, void* d_ws, size_t ws_size,
                              hipStream_t stream) {
    const float* x    = (const float*)d_in[0];
    const float* ea   = (const float*)d_in[1];
    const float* We1  = (const float*)d_in[2];  const float* be1 = (const float*)d_in[3];
    const float* W1a  = (const float*)d_in[4];  const float* b1a = (const float*)d_in[5];
    const float* W1b  = (const float*)d_in[6];  const float* b1b = (const float*)d_in[7];
    const float* We2  = (const float*)d_in[8];  const float* be2 = (const float*)d_in[9];
    const float* W2a  = (const float*)d_in[10]; const float* b2a = (const float*)d_in[11];
    const float* W2b  = (const float*)d_in[12]; const float* b2b = (const float*)d_in[13];
    const float* Wf   = (const float*)d_in[14]; const float* bf  = (const float*)d_in[15];
    const int* eidx   = (const int*)d_in[16];   // [2,E]
    const int* batch  = (const int*)d_in[17];
    const int* src = eidx;
    const int* dst = eidx + N_EDGES;
    float* out = (float*)d_out;

    // workspace layout
    float* aggr = (float*)d_ws;               // N*64
    float* h1   = aggr + (size_t)N_NODES * DIM;
    float* h2   = h1   + (size_t)N_NODES * DIM;
    float* sums = h2   + (size_t)N_NODES * DIM;  // G*64
    float* cnts = sums + (size_t)N_GRAPHS * DIM; // G

    const int nd = N_NODES * DIM;                     // 3.2M
    const int edge_blocks = N_EDGES / 128;            // 12500 (exact)
    const int node_blocks = (N_NODES / 16 + 3) / 4;   // 782

    // layer 1
    zero_kernel<<<(nd + 255) / 256, 256, 0, stream>>>(aggr, nd);
    edge_msg_kernel<<<edge_blocks, 256, 0, stream>>>(x, ea, We1, be1, src, dst, aggr, N_EDGES);
    node_mlp_kernel<<<node_blocks, 128, 0, stream>>>(x, aggr, W1a, b1a, W1b, b1b, h1, N_NODES);

    // layer 2
    zero_kernel<<<(nd + 255) / 256, 256, 0, stream>>>(aggr, nd);
    edge_msg_kernel<<<edge_blocks, 256, 0, stream>>>(h1, ea, We2, be2, src, dst, aggr, N_EDGES);
    node_mlp_kernel<<<node_blocks, 128, 0, stream>>>(h1, aggr, W2a, b2a, W2b, b2b, h2, N_NODES);

    // pool + head
    zero_kernel<<<(N_GRAPHS * DIM + N_GRAPHS + 255) / 256, 256, 0, stream>>>(sums, N_GRAPHS * DIM + N_GRAPHS);
    pool_sum_kernel<<<(nd + 255) / 256, 256, 0, stream>>>(h2, batch, sums, cnts, N_NODES);
    head_kernel<<<1, 128, 0, stream>>>(sums, cnts, Wf, bf, out);
}